// myGAT_10514079941367
// MI455X (gfx1250) — compile-verified
//
#include <hip/hip_runtime.h>
#include <hip/hip_bf16.h>
#include <math.h>

#define HEADS 4
#define HID 128
#define OC (HEADS * HID)      // 512
#define INC 256
#define NEG_SLOPE 0.2f
#define BN_EPS 1e-5f
#define KCHUNK 32             // K depth staged in LDS per double-buffer phase

typedef __attribute__((ext_vector_type(2))) float v2f;
typedef __attribute__((ext_vector_type(8))) float v8f;

// ---------------------------------------------------------------------------
// CDNA5 async global->LDS copy (ASYNCcnt) helpers.
// LDS operand = low 32 bits of the flat shared pointer (ISA 10.2: LDS aperture
// addresses truncate to the LDS byte offset).
// ---------------------------------------------------------------------------
__device__ __forceinline__ void async_load_f128(const float* gsrc, float* ldst) {
  uint32_t loff = (uint32_t)(uintptr_t)ldst;
  asm volatile("global_load_async_to_lds_b128 %0, %1, off"
               :: "v"(loff), "v"(gsrc) : "memory");
}
__device__ __forceinline__ void wait_async0() {
  asm volatile("s_wait_asynccnt 0x0" ::: "memory");
}

// ---------------------------------------------------------------------------
// GEMM: Y[nrows, 512] = X[nrows, K] @ W[K, 512], fp32 WMMA 16x16x4.
// Block = 4 waves = 64 rows x 64 cols. The 64-wide W column tile is shared by
// all 4 waves and staged through LDS in KCHUNK-deep double-buffered slabs via
// global_load_async_to_lds_b128. A-frags stream from global (L2-resident).
// ---------------------------------------------------------------------------
__global__ __launch_bounds__(128) void gemm_f32_wmma(
    const float* __restrict__ X, const float* __restrict__ W,
    float* __restrict__ Y, int nrows, int K) {
  __shared__ float sW[2][KCHUNK * 64];

  const int lane = threadIdx.x & 31;
  const int wave = threadIdx.x >> 5;
  const int tid  = threadIdx.x;
  const int tilesC = OC / 64;                       // 8
  const int rg = blockIdx.x / tilesC;               // row group (4 tiles)
  const int tc = blockIdx.x % tilesC;
  const int col0 = tc * 64;
  const int rowTiles = nrows >> 4;
  const int rt = rg * 4 + wave;
  const bool valid = rt < rowTiles;
  const int row0 = valid ? rt * 16 : 0;             // clamp: stay in barriers

  const int l15  = lane & 15;
  const int koff = (lane >> 4) * 2;
  const float* xrowp = X + (size_t)(row0 + l15) * K + koff;

  const int numChunks = K / KCHUNK;

  // stage chunk 0
#pragma unroll
  for (int i = 0; i < 4; ++i) {
    int f = (tid + i * 128) * 4;                    // float idx in 2048 chunk
    async_load_f128(W + (size_t)(f >> 6) * OC + col0 + (f & 63), &sW[0][f]);
  }

  v8f acc0 = {}, acc1 = {}, acc2 = {}, acc3 = {};
  int buf = 0;
  for (int c = 0; c < numChunks; ++c) {
    wait_async0();
    __syncthreads();
    if (c + 1 < numChunks) {
      const int k0 = (c + 1) * KCHUNK;
#pragma unroll
      for (int i = 0; i < 4; ++i) {
        int f = (tid + i * 128) * 4;
        async_load_f128(W + (size_t)(k0 + (f >> 6)) * OC + col0 + (f & 63),
                        &sW[buf ^ 1][f]);
      }
    }
    const float* wl = sW[buf];
    const int kbase = c * KCHUNK;
#pragma unroll
    for (int kk = 0; kk < KCHUNK; kk += 4) {
      v2f a;
      a.x = xrowp[kbase + kk + 0];
      a.y = xrowp[kbase + kk + 1];
      const float* w0 = wl + (kk + koff) * 64 + l15;
      const float* w1 = w0 + 64;
      v2f b0, b1, b2, b3;
      b0.x = w0[0];  b0.y = w1[0];
      b1.x = w0[16]; b1.y = w1[16];
      b2.x = w0[32]; b2.y = w1[32];
      b3.x = w0[48]; b3.y = w1[48];
      acc0 = __builtin_amdgcn_wmma_f32_16x16x4_f32(false, a, false, b0, (short)0, acc0, false, false);
      acc1 = __builtin_amdgcn_wmma_f32_16x16x4_f32(false, a, false, b1, (short)0, acc1, false, false);
      acc2 = __builtin_amdgcn_wmma_f32_16x16x4_f32(false, a, false, b2, (short)0, acc2, false, false);
      acc3 = __builtin_amdgcn_wmma_f32_16x16x4_f32(false, a, false, b3, (short)0, acc3, false, false);
    }
    buf ^= 1;
  }

  if (valid) {
    const int mbase = row0 + (lane >> 4) * 8;
#pragma unroll
    for (int r = 0; r < 8; ++r) {
      float* yp = Y + (size_t)(mbase + r) * OC + col0 + l15;
      yp[0]  = acc0[r];
      yp[16] = acc1[r];
      yp[32] = acc2[r];
      yp[48] = acc3[r];
    }
  }
}

// ---------------------------------------------------------------------------
// CSR build over incoming edges (done once, reused by both layers).
// ---------------------------------------------------------------------------
__global__ void zero_int_kernel(int* __restrict__ p, int n) {
  int i = blockIdx.x * blockDim.x + threadIdx.x;
  if (i < n) p[i] = 0;
}

__global__ void degree_kernel(const long long* __restrict__ ei, int E, int Etot,
                              int* __restrict__ deg) {
  int e = blockIdx.x * blockDim.x + threadIdx.x;
  if (e >= Etot) return;
  int dst = (e < E) ? (int)ei[E + e] : (e - E);
  atomicAdd(&deg[dst], 1);
}

__global__ __launch_bounds__(256) void scan_kernel(const int* __restrict__ deg,
                                                   int* __restrict__ rowstart,
                                                   int* __restrict__ cursor, int N) {
  __shared__ int part[256];
  const int t = threadIdx.x;
  const int chunk = (N + 255) / 256;
  const int begin = t * chunk;
  const int end = min(begin + chunk, N);
  int s = 0;
  for (int i = begin; i < end; ++i) s += deg[i];
  part[t] = s;
  __syncthreads();
  if (t == 0) {
    int run = 0;
    for (int i = 0; i < 256; ++i) { int tmp = part[i]; part[i] = run; run += tmp; }
    rowstart[N] = run;
  }
  __syncthreads();
  int run = part[t];
  for (int i = begin; i < end; ++i) {
    rowstart[i] = run;
    cursor[i] = run;
    run += deg[i];
  }
}

__global__ void fill_kernel(const long long* __restrict__ ei, int E, int Etot,
                            int* __restrict__ cursor, int* __restrict__ eids) {
  int e = blockIdx.x * blockDim.x + threadIdx.x;
  if (e >= Etot) return;
  int dst = (e < E) ? (int)ei[E + e] : (e - E);
  int p = atomicAdd(&cursor[dst], 1);
  eids[p] = e;
}

// ---------------------------------------------------------------------------
// Softmax state init.
// ---------------------------------------------------------------------------
__global__ void init_softmax(float* __restrict__ lmax, float* __restrict__ denom,
                             int nH) {
  int i = blockIdx.x * blockDim.x + threadIdx.x;
  if (i < nH) { lmax[i] = -__builtin_inff(); denom[i] = 0.f; }
}

__device__ __forceinline__ float atomicMaxF(float* addr, float val) {
  if (val >= 0.f)
    return __int_as_float(atomicMax((int*)addr, __float_as_int(val)));
  else
    return __uint_as_float(atomicMin((unsigned int*)addr, __float_as_uint(val)));
}

// ---------------------------------------------------------------------------
// Pass 1: logits[e,h] = att[h] . leaky_relu(xl[src] + xr[dst]); segment max.
// One wave per edge; 32 lanes * float4 = 128 channels per head.
// ---------------------------------------------------------------------------
__global__ __launch_bounds__(256) void edge_logits_kernel(
    const float* __restrict__ xl, const float* __restrict__ xr,
    const float* __restrict__ att, const long long* __restrict__ ei,
    int E, int Etot, float* __restrict__ logits, float* __restrict__ lmax) {
  const int lane = threadIdx.x & 31;
  const int e = (blockIdx.x * blockDim.x + threadIdx.x) >> 5;
  if (e >= Etot) return;
  const int src = (e < E) ? (int)ei[e]     : (e - E);
  const int dst = (e < E) ? (int)ei[E + e] : (e - E);
  const float4* xlp = (const float4*)(xl + (size_t)src * OC) + lane;
  const float4* xrp = (const float4*)(xr + (size_t)dst * OC) + lane;
  const float4* ap  = (const float4*)att + lane;
#pragma unroll
  for (int h = 0; h < HEADS; ++h) {
    float4 a = ap[h * 32];
    float4 l = xlp[h * 32];
    float4 r = xrp[h * 32];
    float t0 = l.x + r.x; t0 = t0 > 0.f ? t0 : NEG_SLOPE * t0;
    float t1 = l.y + r.y; t1 = t1 > 0.f ? t1 : NEG_SLOPE * t1;
    float t2 = l.z + r.z; t2 = t2 > 0.f ? t2 : NEG_SLOPE * t2;
    float t3 = l.w + r.w; t3 = t3 > 0.f ? t3 : NEG_SLOPE * t3;
    float p = a.x * t0 + a.y * t1 + a.z * t2 + a.w * t3;
#pragma unroll
    for (int off = 16; off > 0; off >>= 1) p += __shfl_down(p, off, 32);
    if (lane == 0) {
      logits[(size_t)e * HEADS + h] = p;
      atomicMaxF(&lmax[dst * HEADS + h], p);
    }
  }
}

// ---------------------------------------------------------------------------
// Pass 2: ex = exp(logit - lmax[dst]); denom[dst] += ex (ex kept in logits).
// ---------------------------------------------------------------------------
__global__ void edge_expsum_kernel(const long long* __restrict__ ei, int E,
                                   int Etot, const float* __restrict__ lmax,
                                   float* __restrict__ logits,
                                   float* __restrict__ denom) {
  int t = blockIdx.x * blockDim.x + threadIdx.x;
  if (t >= Etot * HEADS) return;
  int e = t >> 2, h = t & 3;
  int dst = (e < E) ? (int)ei[E + e] : (e - E);
  float ex = __expf(logits[t] - lmax[dst * HEADS + h]);
  logits[t] = ex;
  atomicAdd(&denom[dst * HEADS + h], ex);
}

// ---------------------------------------------------------------------------
// CSR aggregation, atomic-free: one wave per dst node. Computes
// alpha = ex / (denom[n]+1e-16) inline (denom reciprocals once per node),
// lane 0 streams alpha out to d_out, lanes accumulate alpha*xl[src] in
// registers (4 float4 accs = 16 floats/lane), then fused head-mean + bias +
// BN + ReLU epilogue. Replaces 512 atomics/edge + a whole alpha pass.
// ---------------------------------------------------------------------------
__device__ __forceinline__ void fma4(float s, float4 l, float4& a) {
  a.x = fmaf(s, l.x, a.x); a.y = fmaf(s, l.y, a.y);
  a.z = fmaf(s, l.z, a.z); a.w = fmaf(s, l.w, a.w);
}

__global__ __launch_bounds__(256) void node_aggregate_kernel(
    const float* __restrict__ xl, const long long* __restrict__ ei,
    const int* __restrict__ rowstart, const int* __restrict__ eids,
    const float* __restrict__ logits, const float* __restrict__ denom,
    float* __restrict__ alpha, int E, int N,
    const float* __restrict__ b, const float* __restrict__ g,
    const float* __restrict__ be, const float* __restrict__ m,
    const float* __restrict__ v, float* __restrict__ hout) {
  const int lane = threadIdx.x & 31;
  const int n = (blockIdx.x * blockDim.x + threadIdx.x) >> 5;
  if (n >= N) return;
  const float4 dn = ((const float4*)denom)[n];     // denom[n, 0..3]
  float4 rden;
  rden.x = 1.f / (dn.x + 1e-16f);
  rden.y = 1.f / (dn.y + 1e-16f);
  rden.z = 1.f / (dn.z + 1e-16f);
  rden.w = 1.f / (dn.w + 1e-16f);
  float4 acc0 = {0.f, 0.f, 0.f, 0.f}, acc1 = acc0, acc2 = acc0, acc3 = acc0;
  const int s0 = rowstart[n], s1 = rowstart[n + 1];
  for (int i = s0; i < s1; ++i) {
    const int e = eids[i];
    const int src = (e < E) ? (int)ei[e] : (e - E);
    const float4* xlp = (const float4*)(xl + (size_t)src * OC) + lane;
    const float4 ex = ((const float4*)logits)[e];  // ex[e, 0..3] (broadcast)
    float4 av;
    av.x = ex.x * rden.x;
    av.y = ex.y * rden.y;
    av.z = ex.z * rden.z;
    av.w = ex.w * rden.w;
    if (lane == 0) ((float4*)alpha)[e] = av;
    fma4(av.x, xlp[0],  acc0);
    fma4(av.y, xlp[32], acc1);
    fma4(av.z, xlp[64], acc2);
    fma4(av.w, xlp[96], acc3);
  }
  const float4 bb = ((const float4*)b)[lane];
  const float4 gg = ((const float4*)g)[lane];
  const float4 eb = ((const float4*)be)[lane];
  const float4 mm = ((const float4*)m)[lane];
  const float4 vv = ((const float4*)v)[lane];
  float4 hs;
  hs.x = (acc0.x + acc1.x + acc2.x + acc3.x) * 0.25f + bb.x;
  hs.y = (acc0.y + acc1.y + acc2.y + acc3.y) * 0.25f + bb.y;
  hs.z = (acc0.z + acc1.z + acc2.z + acc3.z) * 0.25f + bb.z;
  hs.w = (acc0.w + acc1.w + acc2.w + acc3.w) * 0.25f + bb.w;
  hs.x = (hs.x - mm.x) * rsqrtf(vv.x + BN_EPS) * gg.x + eb.x;
  hs.y = (hs.y - mm.y) * rsqrtf(vv.y + BN_EPS) * gg.y + eb.y;
  hs.z = (hs.z - mm.z) * rsqrtf(vv.z + BN_EPS) * gg.z + eb.z;
  hs.w = (hs.w - mm.w) * rsqrtf(vv.w + BN_EPS) * gg.w + eb.w;
  hs.x = hs.x > 0.f ? hs.x : 0.f;
  hs.y = hs.y > 0.f ? hs.y : 0.f;
  hs.z = hs.z > 0.f ? hs.z : 0.f;
  hs.w = hs.w > 0.f ? hs.w : 0.f;
  ((float4*)(hout + (size_t)n * HID))[lane] = hs;
}

// ---------------------------------------------------------------------------
// Final: sigmoid(relu(h @ Wlin + blin)). Wave per node, shuffle reduce.
// ---------------------------------------------------------------------------
__global__ __launch_bounds__(256) void final_kernel(
    const float* __restrict__ hin, const float* __restrict__ Wlin,
    const float* __restrict__ blin, float* __restrict__ out, int N) {
  const int lane = threadIdx.x & 31;
  const int n = (blockIdx.x * blockDim.x + threadIdx.x) >> 5;
  if (n >= N) return;
  float4 hv = ((const float4*)(hin + (size_t)n * HID))[lane];
  float4 wv = ((const float4*)Wlin)[lane];
  float p = hv.x * wv.x + hv.y * wv.y + hv.z * wv.z + hv.w * wv.w;
#pragma unroll
  for (int off = 16; off > 0; off >>= 1) p += __shfl_down(p, off, 32);
  if (lane == 0) {
    float x = p + blin[0];
    x = x > 0.f ? x : 0.f;
    out[n] = 1.f / (1.f + __expf(-x));
  }
}

extern "C" void kernel_launch(void* const* d_in, const int* in_sizes, int n_in,
                              void* d_out, int out_size, void* d_ws, size_t ws_size,
                              hipStream_t stream) {
  const float*     x    = (const float*)d_in[0];
  const long long* ei   = (const long long*)d_in[1];  // int64 edge_index [2,E]
  const float*     Wl1  = (const float*)d_in[2];
  const float*     Wr1  = (const float*)d_in[3];
  const float*     att1 = (const float*)d_in[4];
  const float*     b1   = (const float*)d_in[5];
  const float*     Wl2  = (const float*)d_in[6];
  const float*     Wr2  = (const float*)d_in[7];
  const float*     att2 = (const float*)d_in[8];
  const float*     b2   = (const float*)d_in[9];
  const float*     g1   = (const float*)d_in[10];
  const float*     be1  = (const float*)d_in[11];
  const float*     m1   = (const float*)d_in[12];
  const float*     v1   = (const float*)d_in[13];
  const float*     g2   = (const float*)d_in[14];
  const float*     be2  = (const float*)d_in[15];
  const float*     m2   = (const float*)d_in[16];
  const float*     v2   = (const float*)d_in[17];
  const float*     Wlin = (const float*)d_in[18];
  const float*     blin = (const float*)d_in[19];

  const int N    = in_sizes[0] / INC;   // 10000
  const int E    = in_sizes[1] / 2;     // 320000
  const int Etot = E + N;               // self loops appended

  // Workspace: floats first, then int CSR arrays. xl/xr (20.5MB each) stay L2-resident.
  float* ws     = (float*)d_ws;
  float* xl     = ws;                              // N*OC
  float* xr     = xl + (size_t)N * OC;             // N*OC
  float* hbuf   = xr + (size_t)N * OC;             // N*HID
  float* logits = hbuf + (size_t)N * HID;          // Etot*HEADS
  float* lmax   = logits + (size_t)Etot * HEADS;   // N*HEADS
  float* denom  = lmax + (size_t)N * HEADS;        // N*HEADS
  int*   deg      = (int*)(denom + (size_t)N * HEADS); // N
  int*   rowstart = deg + N;                            // N+1
  int*   cursor   = rowstart + (N + 1);                 // N
  int*   eids     = cursor + N;                         // Etot

  float* out_pred = (float*)d_out;                 // N
  float* alpha1   = out_pred + N;                  // Etot*HEADS
  float* alpha2   = alpha1 + (size_t)Etot * HEADS; // Etot*HEADS

  const int tilesC = OC / 64;
  const int gemmBlocks = ((N / 16 + 3) / 4) * tilesC;
  const int smBlocks = (N * HEADS + 255) / 256;
  const int edgeWaveBlocks = (Etot + 7) / 8;            // 8 waves / 256-thr block
  const int edgeThrBlocks  = (Etot * HEADS + 255) / 256;
  const int edgeBlocks1 = (Etot + 255) / 256;
  const int nodeWaveBlocks = (N + 7) / 8;

  // ---------------- CSR build (once, shared by both layers) ----------------
  zero_int_kernel<<<(N + 255) / 256, 256, 0, stream>>>(deg, N);
  degree_kernel<<<edgeBlocks1, 256, 0, stream>>>(ei, E, Etot, deg);
  scan_kernel<<<1, 256, 0, stream>>>(deg, rowstart, cursor, N);
  fill_kernel<<<edgeBlocks1, 256, 0, stream>>>(ei, E, Etot, cursor, eids);

  // ---------------- Layer 1 ----------------
  gemm_f32_wmma<<<gemmBlocks, 128, 0, stream>>>(x, Wl1, xl, N, INC);
  gemm_f32_wmma<<<gemmBlocks, 128, 0, stream>>>(x, Wr1, xr, N, INC);
  init_softmax<<<smBlocks, 256, 0, stream>>>(lmax, denom, N * HEADS);
  edge_logits_kernel<<<edgeWaveBlocks, 256, 0, stream>>>(xl, xr, att1, ei, E, Etot, logits, lmax);
  edge_expsum_kernel<<<edgeThrBlocks, 256, 0, stream>>>(ei, E, Etot, lmax, logits, denom);
  node_aggregate_kernel<<<nodeWaveBlocks, 256, 0, stream>>>(
      xl, ei, rowstart, eids, logits, denom, alpha1, E, N, b1, g1, be1, m1, v1, hbuf);

  // ---------------- Layer 2 ----------------
  gemm_f32_wmma<<<gemmBlocks, 128, 0, stream>>>(hbuf, Wl2, xl, N, HID);
  gemm_f32_wmma<<<gemmBlocks, 128, 0, stream>>>(hbuf, Wr2, xr, N, HID);
  init_softmax<<<smBlocks, 256, 0, stream>>>(lmax, denom, N * HEADS);
  edge_logits_kernel<<<edgeWaveBlocks, 256, 0, stream>>>(xl, xr, att2, ei, E, Etot, logits, lmax);
  edge_expsum_kernel<<<edgeThrBlocks, 256, 0, stream>>>(ei, E, Etot, lmax, logits, denom);
  node_aggregate_kernel<<<nodeWaveBlocks, 256, 0, stream>>>(
      xl, ei, rowstart, eids, logits, denom, alpha2, E, N, b2, g2, be2, m2, v2, hbuf);

  // ---------------- Head ----------------
  final_kernel<<<(N + 7) / 8, 256, 0, stream>>>(hbuf, Wlin, blin, out_pred, N);
}